// FNOnd_53961969107531
// MI455X (gfx1250) — compile-verified
//
#include <hip/hip_runtime.h>

// ---------------------------------------------------------------------------
// FNO forward for MI455X (gfx1250).
// Truncated-mode DFTs + spectral mix + bypass conv are all mapped onto
// V_WMMA_F32_16X16X4_F32 (fp32 in / fp32 acc).  Inverse column DFT is fused
// with the bypass conv1x1 + GELU through a 64KB LDS tile to cut one full
// 134MB activation round trip per block (bandwidth-bound workload).
// Twiddles come straight from V_SIN_F32/V_COS_F32 (input in revolutions:
// sin(2*pi*ph/256) == v_sin(ph/256)) -- no libm range reduction.
// ---------------------------------------------------------------------------

#define NBLK   4
#define B_     8
#define C_     64          // WIDTH
#define H_     256
#define W_     256
#define HW_    65536
#define M_     16          // kept modes per axis
#define INV256 0.00390625f               // 1/256 (phase -> revolutions)
#define INV16  0.0625f                   // ortho 1/sqrt(256)
#define TWO16  0.125f                    // 2/16

typedef __attribute__((ext_vector_type(2))) float v2f;
typedef __attribute__((ext_vector_type(8))) float v8f;

static __device__ __forceinline__ v8f wmma4(v2f a, v2f b, v8f c) {
  // D = A(16x4) * B(4x16) + C(16x16), fp32
  return __builtin_amdgcn_wmma_f32_16x16x4_f32(false, a, false, b, (short)0, c,
                                               false, false);
}
static __device__ __forceinline__ v8f vzero() {
  v8f z = {0.f, 0.f, 0.f, 0.f, 0.f, 0.f, 0.f, 0.f};
  return z;
}
// sin/cos of 2*pi*ph/256, ph integer phase; hardware unit, no reduction.
static __device__ __forceinline__ void tw(int ph, float& s, float& c) {
  float r = (float)ph * INV256;
  s = __builtin_amdgcn_sinf(r);
  c = __builtin_amdgcn_cosf(r);
}

// ---------------------------------------------------------------------------
// K1: lift.  out[b,c,h,w] = lw[c,0]*x + lw[c,1]*g1(h) + lw[c,2]*g2(w) + lb[c]
// ---------------------------------------------------------------------------
__global__ void k_lift(const float* __restrict__ x, const float* __restrict__ lw,
                       const float* __restrict__ lb, float* __restrict__ skip) {
  int idx = blockIdx.x * blockDim.x + threadIdx.x;   // [0, B*H*W)
  int w = idx & 255;
  int h = (idx >> 8) & 255;
  int b = idx >> 16;
  float xv = x[idx];
  float g1 = (float)h * (1.0f / 255.0f);
  float g2 = (float)w * (1.0f / 255.0f);
  size_t base = ((size_t)b * C_) * HW_ + (size_t)h * W_ + (size_t)w;
  for (int c = 0; c < C_; ++c) {
    float v = lw[c * 3 + 0] * xv + lw[c * 3 + 1] * g1 + lw[c * 3 + 2] * g2 + lb[c];
    skip[base + (size_t)c * HW_] = v;
  }
}

// ---------------------------------------------------------------------------
// K2: forward row DFT.  T[kx,w] = sum_h x[h,w] * e^{-i 2pi kx h/256} / 16
// One wave per (b*c, w-tile).  M=kx(16), K=h(256), N=w(16).
// ---------------------------------------------------------------------------
__global__ void k_fwd_row(const float* __restrict__ x, float* __restrict__ Tr,
                          float* __restrict__ Ti) {
  int gw   = (blockIdx.x * blockDim.x + threadIdx.x) >> 5;   // wave id
  int lane = threadIdx.x & 31;
  int wt = gw & 15;
  int bc = gw >> 4;                                          // [0, 512)
  int m    = lane & 15;
  int koff = (lane >> 4) << 1;
  int mb8  = (lane >> 4) << 3;
  const float* xp = x + (size_t)bc * HW_;
  int w0 = wt * 16;
  v8f accR = vzero(), accI = vzero();
  for (int k0 = 0; k0 < 256; k0 += 4) {
    int h0 = k0 + koff;
    // A = e^{-i*2pi*kx*h/256}/16 ; kx = m, h = h0, h0+1
    float s0, c0, s1, c1;
    tw((m * h0) & 255, s0, c0);
    tw((m * (h0 + 1)) & 255, s1, c1);
    v2f ar, ai;
    ar.x =  c0 * INV16;  ar.y =  c1 * INV16;
    ai.x = -s0 * INV16;  ai.y = -s1 * INV16;
    // B = x[h][w] ; k = h, n = w0+m
    v2f bx;
    bx.x = xp[(size_t)h0 * W_ + w0 + m];
    bx.y = xp[(size_t)(h0 + 1) * W_ + w0 + m];
    accR = wmma4(ar, bx, accR);
    accI = wmma4(ai, bx, accI);
  }
  float* trp = Tr + (size_t)bc * (M_ * W_);
  float* tip = Ti + (size_t)bc * (M_ * W_);
  for (int r = 0; r < 8; ++r) {
    int kx = mb8 + r;
    trp[kx * W_ + w0 + m] = accR[r];
    tip[kx * W_ + w0 + m] = accI[r];
  }
}

// ---------------------------------------------------------------------------
// K3: forward column DFT.  xf[kx,ky] = sum_w T[kx,w] * e^{-i 2pi ky w/256}/16
// One wave per (b*c).  M=kx(16), K=w(256), N=ky(16).  Complex x complex.
// ---------------------------------------------------------------------------
__global__ void k_fwd_col(const float* __restrict__ Tr, const float* __restrict__ Ti,
                          float* __restrict__ xfr, float* __restrict__ xfi) {
  int bc   = (blockIdx.x * blockDim.x + threadIdx.x) >> 5;   // [0, 512)
  int lane = threadIdx.x & 31;
  int m    = lane & 15;
  int koff = (lane >> 4) << 1;
  int mb8  = (lane >> 4) << 3;
  const float* trp = Tr + (size_t)bc * (M_ * W_);
  const float* tip = Ti + (size_t)bc * (M_ * W_);
  v8f accR = vzero(), accI = vzero();
  for (int k0 = 0; k0 < 256; k0 += 4) {
    int w_ = k0 + koff;
    v2f ar, ai;
    ar.x = trp[m * W_ + w_];     ar.y = trp[m * W_ + w_ + 1];
    ai.x = tip[m * W_ + w_];     ai.y = tip[m * W_ + w_ + 1];
    v2f nai = -ai;
    // B = e^{-i*2pi*ky*w/256}/16 ; k = w, n = ky = m
    float s0, c0, s1, c1;
    tw((m * w_) & 255, s0, c0);
    tw((m * (w_ + 1)) & 255, s1, c1);
    v2f gr, gi;
    gr.x =  c0 * INV16;  gr.y =  c1 * INV16;
    gi.x = -s0 * INV16;  gi.y = -s1 * INV16;
    accR = wmma4(ar,  gr, accR);
    accR = wmma4(nai, gi, accR);   // - Ti*Gi
    accI = wmma4(ar,  gi, accI);
    accI = wmma4(ai,  gr, accI);
  }
  float* xr = xfr + (size_t)bc * (M_ * M_);
  float* xi = xfi + (size_t)bc * (M_ * M_);
  for (int r = 0; r < 8; ++r) {
    int kx = mb8 + r;
    xr[kx * M_ + m] = accR[r];
    xi[kx * M_ + m] = accI[r];
  }
}

// ---------------------------------------------------------------------------
// K4: spectral mix.  out[b,o,md] = sum_i xf[b,i,md] * (wr+i*wi)[i,o,md]
// One wave per (mode, o-tile).  M=batch(8 used of 16), K=i(64), N=o(16).
// ---------------------------------------------------------------------------
__global__ void k_spec(const float* __restrict__ xfr, const float* __restrict__ xfi,
                       const float* __restrict__ wr, const float* __restrict__ wi,
                       float* __restrict__ ofr, float* __restrict__ ofi) {
  int gw   = (blockIdx.x * blockDim.x + threadIdx.x) >> 5;   // [0, 1024)
  int lane = threadIdx.x & 31;
  int ot = gw & 3;
  int md = gw >> 2;                                          // [0, 256)
  int m    = lane & 15;
  int koff = (lane >> 4) << 1;
  int o0 = ot * 16;
  v8f accR = vzero(), accI = vzero();
  for (int k0 = 0; k0 < C_; k0 += 4) {
    int i0 = k0 + koff;
    v2f ar, ai;
    ar.x = (m < B_) ? xfr[((size_t)(m * C_ + i0)) * 256 + md] : 0.f;
    ar.y = (m < B_) ? xfr[((size_t)(m * C_ + i0 + 1)) * 256 + md] : 0.f;
    ai.x = (m < B_) ? xfi[((size_t)(m * C_ + i0)) * 256 + md] : 0.f;
    ai.y = (m < B_) ? xfi[((size_t)(m * C_ + i0 + 1)) * 256 + md] : 0.f;
    v2f nai = -ai;
    v2f bwr, bwi;
    bwr.x = wr[((size_t)(i0 * C_ + o0 + m)) * 256 + md];
    bwr.y = wr[((size_t)((i0 + 1) * C_ + o0 + m)) * 256 + md];
    bwi.x = wi[((size_t)(i0 * C_ + o0 + m)) * 256 + md];
    bwi.y = wi[((size_t)((i0 + 1) * C_ + o0 + m)) * 256 + md];
    accR = wmma4(ar,  bwr, accR);
    accR = wmma4(nai, bwi, accR);
    accI = wmma4(ar,  bwi, accI);
    accI = wmma4(ai,  bwr, accI);
  }
  if (lane < 16) {                      // rows 0..7 = real batches
    for (int r = 0; r < B_; ++r) {
      ofr[((size_t)(r * C_ + o0 + m)) * 256 + md] = accR[r];
      ofi[((size_t)(r * C_ + o0 + m)) * 256 + md] = accI[r];
    }
  }
}

// ---------------------------------------------------------------------------
// K5: inverse row transform.  U[h,ky] = sum_kx S[kx,ky] e^{+i 2pi kx h/256}/16
// One wave per (b*c, h-tile).  M=h(16), K=kx(16), N=ky(16).
// ---------------------------------------------------------------------------
__global__ void k_inv_row(const float* __restrict__ ofr, const float* __restrict__ ofi,
                          float* __restrict__ Ur, float* __restrict__ Ui) {
  int gw   = (blockIdx.x * blockDim.x + threadIdx.x) >> 5;   // [0, 8192)
  int lane = threadIdx.x & 31;
  int ht = gw & 15;
  int bc = gw >> 4;                                          // [0, 512)
  int m    = lane & 15;
  int koff = (lane >> 4) << 1;
  int mb8  = (lane >> 4) << 3;
  int h0 = ht * 16;
  const float* sr = ofr + (size_t)bc * (M_ * M_);
  const float* si = ofi + (size_t)bc * (M_ * M_);
  v8f accR = vzero(), accI = vzero();
  for (int k0 = 0; k0 < M_; k0 += 4) {
    int kx0 = k0 + koff;
    // A = e^{+i*2pi*kx*h/256}/16 ; h = h0+m, kx = kx0, kx0+1
    float s0, c0, s1, c1;
    tw((kx0 * (h0 + m)) & 255, s0, c0);
    tw(((kx0 + 1) * (h0 + m)) & 255, s1, c1);
    v2f er, ei;
    er.x = c0 * INV16;  er.y = c1 * INV16;
    ei.x = s0 * INV16;  ei.y = s1 * INV16;
    v2f nei = -ei;
    // B = S[kx][ky] ; k = kx, n = ky = m
    v2f bsr, bsi;
    bsr.x = sr[kx0 * M_ + m];       bsr.y = sr[(kx0 + 1) * M_ + m];
    bsi.x = si[kx0 * M_ + m];       bsi.y = si[(kx0 + 1) * M_ + m];
    accR = wmma4(er,  bsr, accR);
    accR = wmma4(nei, bsi, accR);
    accI = wmma4(er,  bsi, accI);
    accI = wmma4(ei,  bsr, accI);
  }
  float* urp = Ur + (size_t)bc * (H_ * M_);
  float* uip = Ui + (size_t)bc * (H_ * M_);
  for (int r = 0; r < 8; ++r) {
    int h = h0 + mb8 + r;
    urp[h * M_ + m] = accR[r];
    uip[h * M_ + m] = accI[r];
  }
}

// ---------------------------------------------------------------------------
// K6: fused inverse column transform + bypass conv1x1 + bias + GELU.
// One 256-thread workgroup per (b, 16x16 pixel tile).
//   Phase 1 (stage D):  xs[c][hh*16+ww] = sum_ky Ur*Cw + Ui*Dw   (per channel)
//   Phase 2:            out = gelu(xs[o][p] + sum_i bw[o,i]*x_in[i,p] + bb[o])
// ---------------------------------------------------------------------------
__global__ void __launch_bounds__(256) k_bypass(
    const float* __restrict__ Ur, const float* __restrict__ Ui,
    const float* __restrict__ xin, const float* __restrict__ bw,
    const float* __restrict__ bb, float* __restrict__ xout) {
  __shared__ float xs[C_][256];          // 64 KB: spectral activations for tile
  int wv   = threadIdx.x >> 5;
  int lane = threadIdx.x & 31;
  int m    = lane & 15;
  int koff = (lane >> 4) << 1;
  int mb8  = (lane >> 4) << 3;
  int wt = blockIdx.x & 15;
  int ht = (blockIdx.x >> 4) & 15;
  int b  = blockIdx.x >> 8;
  int h0 = ht * 16, w0 = wt * 16;

  // ---- phase 1: inverse column DFT, 8 channels per wave ----
  for (int cc = 0; cc < 8; ++cc) {
    int c = wv * 8 + cc;
    const float* urp = Ur + ((size_t)(b * C_ + c)) * (H_ * M_);
    const float* uip = Ui + ((size_t)(b * C_ + c)) * (H_ * M_);
    v8f acc = vzero();
    for (int k0 = 0; k0 < M_; k0 += 4) {
      int ky0 = k0 + koff;
      int ky1 = ky0 + 1;
      // A = U[h][ky] ; h = h0+m
      v2f aur, aui;
      aur.x = urp[(h0 + m) * M_ + ky0];   aur.y = urp[(h0 + m) * M_ + ky1];
      aui.x = uip[(h0 + m) * M_ + ky0];   aui.y = uip[(h0 + m) * M_ + ky1];
      // B: Cw[ky][w] = (ky==0 ? 1/16 : 2cos/16), Dw[ky][w] = (ky==0 ? 0 : -2sin/16)
      int wg = w0 + m;
      float s0, c0, s1, c1;
      tw((ky0 * wg) & 255, s0, c0);
      tw((ky1 * wg) & 255, s1, c1);
      v2f bc_, bd_;
      bc_.x = (ky0 == 0) ? INV16 :  c0 * TWO16;
      bc_.y = (ky1 == 0) ? INV16 :  c1 * TWO16;
      bd_.x = (ky0 == 0) ? 0.f   : -s0 * TWO16;
      bd_.y = (ky1 == 0) ? 0.f   : -s1 * TWO16;
      acc = wmma4(aur, bc_, acc);
      acc = wmma4(aui, bd_, acc);
    }
    for (int r = 0; r < 8; ++r)
      xs[c][(mb8 + r) * 16 + m] = acc[r];
  }
  __syncthreads();

  // ---- phase 2: bypass conv1x1 + bias + spectral add + GELU ----
  const float* xbase = xin + ((size_t)b * C_) * HW_;
  for (int tt = 0; tt < 8; ++tt) {
    int t  = wv * 8 + tt;
    int ot = t >> 4;                  // o tile [0,4)
    int pt = t & 15;                  // pixel row within tile -> h0+pt
    int o0 = ot * 16;
    const float* xrow = xbase + (size_t)(h0 + pt) * W_ + w0;
    v8f acc = vzero();
    for (int k0 = 0; k0 < C_; k0 += 4) {
      int i0 = k0 + koff;
      __builtin_prefetch(xrow + (size_t)(i0 + 8) * HW_, 0, 0);
      v2f a_, b_;
      a_.x = bw[(o0 + m) * C_ + i0];
      a_.y = bw[(o0 + m) * C_ + i0 + 1];
      b_.x = xrow[(size_t)i0 * HW_ + m];
      b_.y = xrow[(size_t)(i0 + 1) * HW_ + m];
      acc = wmma4(a_, b_, acc);
    }
    for (int r = 0; r < 8; ++r) {
      int o = o0 + mb8 + r;
      float v = acc[r] + bb[o] + xs[o][pt * 16 + m];
      v = 0.5f * v * (1.0f + erff(v * 0.70710678118654752f));   // exact GELU
      xout[((size_t)(b * C_ + o)) * HW_ + (size_t)(h0 + pt) * W_ + w0 + m] = v;
    }
  }
}

// ---------------------------------------------------------------------------
// K7: residual skip + projection 64 -> 1
// ---------------------------------------------------------------------------
__global__ void k_proj(const float* __restrict__ act, const float* __restrict__ skip,
                       const float* __restrict__ pw, const float* __restrict__ pb,
                       float* __restrict__ out) {
  int idx = blockIdx.x * blockDim.x + threadIdx.x;  // [0, B*H*W)
  int hw = idx & (HW_ - 1);
  int b  = idx >> 16;
  float s = pb[0];
  size_t base = ((size_t)b * C_) * HW_ + hw;
  for (int c = 0; c < C_; ++c)
    s += pw[c] * (act[base + (size_t)c * HW_] + skip[base + (size_t)c * HW_]);
  out[idx] = s;
}

// ---------------------------------------------------------------------------
extern "C" void kernel_launch(void* const* d_in, const int* in_sizes, int n_in,
                              void* d_out, int out_size, void* d_ws, size_t ws_size,
                              hipStream_t stream) {
  (void)in_sizes; (void)n_in; (void)out_size; (void)ws_size;
  const float* x       = (const float*)d_in[0];
  const float* lift_w  = (const float*)d_in[1];
  const float* lift_b  = (const float*)d_in[2];
  const float* spec_wr = (const float*)d_in[3];
  const float* spec_wi = (const float*)d_in[4];
  const float* byp_w   = (const float*)d_in[5];
  const float* byp_b   = (const float*)d_in[6];
  const float* proj_w  = (const float*)d_in[7];
  const float* proj_b  = (const float*)d_in[8];

  // workspace layout (floats); total ~438 MB
  float* ws = (float*)d_ws;
  const size_t ACT = (size_t)B_ * C_ * HW_;          // 33,554,432
  const size_t TSZ = (size_t)B_ * C_ * M_ * W_;      //  2,097,152
  const size_t XSZ = (size_t)B_ * C_ * M_ * M_;      //    131,072
  const size_t USZ = (size_t)B_ * C_ * H_ * M_;      //  2,097,152
  float* skip = ws;
  float* act0 = skip + ACT;
  float* act1 = act0 + ACT;
  float* Tr   = act1 + ACT;
  float* Ti   = Tr + TSZ;
  float* xfr  = Ti + TSZ;
  float* xfi  = xfr + XSZ;
  float* ofr  = xfi + XSZ;
  float* ofi  = ofr + XSZ;
  float* Ur   = ofi + XSZ;
  float* Ui   = Ur + USZ;

  k_lift<<<2048, 256, 0, stream>>>(x, lift_w, lift_b, skip);

  const float* cur = skip;
  float* pp[2] = {act0, act1};
  for (int blk = 0; blk < NBLK; ++blk) {
    float* nxt = pp[blk & 1];
    k_fwd_row<<<2048, 128, 0, stream>>>(cur, Tr, Ti);                   // 8192 waves
    k_fwd_col<<<128, 128, 0, stream>>>(Tr, Ti, xfr, xfi);               //  512 waves
    k_spec<<<256, 128, 0, stream>>>(xfr, xfi,
                                    spec_wr + (size_t)blk * C_ * C_ * M_ * M_,
                                    spec_wi + (size_t)blk * C_ * C_ * M_ * M_,
                                    ofr, ofi);                          // 1024 waves
    k_inv_row<<<2048, 128, 0, stream>>>(ofr, ofi, Ur, Ui);              // 8192 waves
    k_bypass<<<2048, 256, 0, stream>>>(Ur, Ui, cur,
                                       byp_w + (size_t)blk * C_ * C_,
                                       byp_b + (size_t)blk * C_, nxt);
    cur = nxt;
  }
  k_proj<<<2048, 256, 0, stream>>>(cur, skip, proj_w, proj_b, (float*)d_out);
}